// PeriodicPaddingLayer_34093450396641
// MI455X (gfx1250) — compile-verified
//
#include <hip/hip_runtime.h>

// Circular (wrap) padding: (64,256,256,16) f32 -> (64,260,260,16) f32, pad=2.
// Pure data movement: ~524 MiB total traffic -> ~23 us at 23.3 TB/s HBM.
// Strategy: async global->LDS->global B128 DMA path (CDNA5 ASYNCcnt ops),
// no VGPR data round-trip, fully coalesced, zero-divergence index math,
// non-temporal cache policy (streamed data, output never re-read).

#define PADP 2
#define NB   64
#define HI   256
#define WI   256
#define NC   16
#define HO   260
#define WO   260

#define THREADS 256
#define ELEMS   8                                   // float4 per thread (8 async B128 in flight)
#define VEC4_PER_BLOCK (THREADS * ELEMS)            // 2048 float4 = 32 KB LDS
#define OUT_VEC4 (NB * HO * WO * (NC / 4))          // 17,305,600
#define NBLOCKS (OUT_VEC4 / VEC4_PER_BLOCK)         // 8,450 exactly (no tail)

__global__ __launch_bounds__(THREADS)
void periodic_pad_async_kernel(const float* __restrict__ in,
                               float* __restrict__ out)
{
    __shared__ float4 stage[VEC4_PER_BLOCK];        // 32 KB staging buffer

    const unsigned base = blockIdx.x * (unsigned)VEC4_PER_BLOCK;

    unsigned      ldsByte[ELEMS];
    const float4* srcp[ELEMS];
    float4*       dstp[ELEMS];

#pragma unroll
    for (int e = 0; e < ELEMS; ++e) {
        // Global output float4 index (coalesced across lanes).
        unsigned idx = base + (unsigned)(e * THREADS) + threadIdx.x;

        // Decompose: idx = (((b*HO + y)*WO + x)*4 + c4)
        unsigned c4 = idx & 3u;
        unsigned p  = idx >> 2;           // output pixel index
        unsigned x  = p % (unsigned)WO;   // magic-number div (WO=260)
        unsigned t  = p / (unsigned)WO;
        unsigned y  = t % (unsigned)HO;
        unsigned b  = t / (unsigned)HO;

        // Circular wrap: (x - 2) mod 256 == (x + 254) & 255  (power-of-two!)
        unsigned xin = (x + (unsigned)(WI - PADP)) & (unsigned)(WI - 1);
        unsigned yin = (y + (unsigned)(HI - PADP)) & (unsigned)(HI - 1);

        // Input float4 index: ((b*256 + yin)*256 + xin)*4 + c4
        unsigned s = ((((b << 8) + yin) << 8) + xin) * 4u + c4;

        srcp[e] = (const float4*)in + s;
        dstp[e] = (float4*)out + idx;
        // Group-segment-relative LDS byte address (low 32 bits of flat addr);
        // taking the address also keeps `stage` allocated.
        ldsByte[e] = (unsigned)(unsigned long long)
                         &stage[e * THREADS + (int)threadIdx.x];
    }

    // Phase 1: 8 async B128 DMA loads per lane, global -> LDS (ASYNCcnt += 8).
    // Non-temporal: input is streamed once (256 MiB > 192 MB L2).
#pragma unroll
    for (int e = 0; e < ELEMS; ++e) {
        asm volatile("global_load_async_to_lds_b128 %0, %1, off th:TH_LOAD_NT"
                     :
                     : "v"(ldsByte[e]), "v"(srcp[e])
                     : "memory");
    }

    // Wait for this wave's async loads to land in LDS before reading them back.
    asm volatile("s_wait_asynccnt 0" ::: "memory");

    // Phase 2: 8 async B128 DMA stores per lane, LDS -> global.
    // Non-temporal: output is never re-read; don't thrash L2 with it.
#pragma unroll
    for (int e = 0; e < ELEMS; ++e) {
        asm volatile("global_store_async_from_lds_b128 %0, %1, off th:TH_STORE_NT"
                     :
                     : "v"(dstp[e]), "v"(ldsByte[e])
                     : "memory");
    }

    // No trailing wait needed: S_ENDPGM performs an implicit wait-idle.
}

extern "C" void kernel_launch(void* const* d_in, const int* in_sizes, int n_in,
                              void* d_out, int out_size, void* d_ws, size_t ws_size,
                              hipStream_t stream)
{
    const float* in  = (const float*)d_in[0];
    float*       out = (float*)d_out;

    periodic_pad_async_kernel<<<NBLOCKS, THREADS, 0, stream>>>(in, out);
}